// PointNetShapeServoKp3_32014686224761
// MI455X (gfx1250) — compile-verified
//
#include <hip/hip_runtime.h>

typedef float v2f __attribute__((ext_vector_type(2)));
typedef float v8f __attribute__((ext_vector_type(8)));

#define EPSV 1e-5f

// ---------------------------------------------------------------------------
// transpose (B,6,N) -> pts (B,N,6) and xyz0 (B,N,3)
// ---------------------------------------------------------------------------
__global__ __launch_bounds__(256) void k_transpose(const float* __restrict__ in,
                                                   float* __restrict__ pts,
                                                   float* __restrict__ xyz0) {
  const int N = 2048;
  int idx = blockIdx.x * 256 + threadIdx.x;
  if (idx >= 16 * 6 * N) return;
  int n = idx % N;
  int c = (idx / N) % 6;
  int b = idx / (6 * N);
  float v = in[idx];
  pts[((size_t)b * N + n) * 6 + c] = v;
  if (c < 3) xyz0[((size_t)b * N + n) * 3 + c] = v;
}

// ---------------------------------------------------------------------------
// density: invden[b,n] = 1 / ( mean_j exp(-d2/(2 bw^2)) / (2.5 bw) )
// one block = (batch, 256-row chunk); all N points staged in LDS
// ---------------------------------------------------------------------------
__global__ __launch_bounds__(256) void k_density(const float* __restrict__ xyz,
                                                 float* __restrict__ invden,
                                                 int N, float bw) {
  __shared__ float sp[2048 * 3];
  int b = blockIdx.x;
  int tid = threadIdx.x;
  for (int i = tid; i < N * 3; i += 256) sp[i] = xyz[(size_t)b * N * 3 + i];
  __syncthreads();
  int row = blockIdx.y * 256 + tid;
  if (row >= N) return;
  float px = sp[row * 3], py = sp[row * 3 + 1], pz = sp[row * 3 + 2];
  float cc = -1.f / (2.f * bw * bw);
  float s = 0.f;
  for (int j = 0; j < N; ++j) {
    float dx = sp[j * 3] - px, dy = sp[j * 3 + 1] - py, dz = sp[j * 3 + 2] - pz;
    s += __expf(cc * (dx * dx + dy * dy + dz * dz));
  }
  float den = (s / (float)N) / (2.5f * bw);
  invden[(size_t)b * N + row] = 1.f / den;
}

// ---------------------------------------------------------------------------
// farthest point sampling: sequential npoint iterations, block argmax
// ---------------------------------------------------------------------------
__global__ __launch_bounds__(256) void k_fps(const float* __restrict__ xyz, int N,
                                             int npoint, int* __restrict__ out) {
  int b = blockIdx.x, tid = threadIdx.x;
  __shared__ float dist[2048];
  __shared__ float rv[256];
  __shared__ int ri[256];
  const float* base = xyz + (size_t)b * N * 3;
  for (int i = tid; i < N; i += 256) dist[i] = 1e10f;
  __syncthreads();
  int far = 0;
  for (int t = 0; t < npoint; ++t) {
    if (tid == 0) out[b * npoint + t] = far;
    float cx = base[far * 3], cy = base[far * 3 + 1], cz = base[far * 3 + 2];
    float bv = -1.f;
    int bi = 0;
    for (int i = tid; i < N; i += 256) {
      float dx = base[i * 3] - cx, dy = base[i * 3 + 1] - cy, dz = base[i * 3 + 2] - cz;
      float d = dx * dx + dy * dy + dz * dz;
      float nd = fminf(dist[i], d);
      dist[i] = nd;
      if (nd > bv) { bv = nd; bi = i; }
    }
    rv[tid] = bv;
    ri[tid] = bi;
    __syncthreads();
    for (int s = 128; s > 0; s >>= 1) {
      if (tid < s) {
        if (rv[tid + s] > rv[tid] || (rv[tid + s] == rv[tid] && ri[tid + s] < ri[tid])) {
          rv[tid] = rv[tid + s];
          ri[tid] = ri[tid + s];
        }
      }
      __syncthreads();
    }
    far = ri[0];
    __syncthreads();
  }
}

// ---------------------------------------------------------------------------
// kNN (register insertion sort) + new_xyz + density ratio per (b,s)
// ---------------------------------------------------------------------------
template <int NS>
__global__ __launch_bounds__(256) void k_knn(const float* __restrict__ xyz, int N,
                                             const int* __restrict__ fps, int S,
                                             const float* __restrict__ invden,
                                             float* __restrict__ nx,
                                             int* __restrict__ knn,
                                             float* __restrict__ ratio, int total) {
  int gid = blockIdx.x * 256 + threadIdx.x;
  if (gid >= total) return;
  int b = gid / S, s = gid % S;
  const float* base = xyz + (size_t)b * N * 3;
  int ci = fps[b * S + s];
  float cx = base[ci * 3], cy = base[ci * 3 + 1], cz = base[ci * 3 + 2];
  nx[(size_t)gid * 3 + 0] = cx;
  nx[(size_t)gid * 3 + 1] = cy;
  nx[(size_t)gid * 3 + 2] = cz;
  float bd[NS];
  int bi[NS];
#pragma unroll
  for (int t = 0; t < NS; ++t) { bd[t] = 3.4e38f; bi[t] = 0; }
  for (int j = 0; j < N; ++j) {
    float dx = base[j * 3] - cx, dy = base[j * 3 + 1] - cy, dz = base[j * 3 + 2] - cz;
    float d = dx * dx + dy * dy + dz * dz;
    if (d < bd[NS - 1]) {
      bd[NS - 1] = d;
      bi[NS - 1] = j;
#pragma unroll
      for (int t = NS - 1; t > 0; --t) {
        if (bd[t] < bd[t - 1]) {
          float tv = bd[t]; bd[t] = bd[t - 1]; bd[t - 1] = tv;
          int ti = bi[t]; bi[t] = bi[t - 1]; bi[t - 1] = ti;
        }
      }
    }
  }
  float gmax = 0.f;
  float gv[NS];
#pragma unroll
  for (int t = 0; t < NS; ++t) {
    gv[t] = invden[(size_t)b * N + bi[t]];
    gmax = fmaxf(gmax, gv[t]);
  }
#pragma unroll
  for (int t = 0; t < NS; ++t) {
    knn[(size_t)gid * NS + t] = bi[t];
    ratio[(size_t)gid * NS + t] = gv[t] / gmax;
  }
}

// ---------------------------------------------------------------------------
// gather: build A rows [xyz_norm(3) | feats(FC) | pad0] and Win rows [norm(3)|0]
// ---------------------------------------------------------------------------
__global__ __launch_bounds__(256) void k_gather(const float* __restrict__ xyz, int N,
                                                const float* __restrict__ feats, int FC,
                                                const float* __restrict__ nx, int S, int NS,
                                                const int* __restrict__ knn,
                                                float* __restrict__ A, int lda,
                                                float* __restrict__ Win, long total) {
  long idx = (long)blockIdx.x * 256 + threadIdx.x;
  if (idx >= total) return;
  long r = idx / lda;
  int c = (int)(idx % lda);
  long rs = r / NS;
  int s = (int)(rs % S);
  int b = (int)(rs / S);
  int j = knn[r];
  float v;
  if (c < 3) {
    v = xyz[((size_t)b * N + j) * 3 + c] - nx[((size_t)b * S + s) * 3 + c];
    Win[r * 4 + c] = v;
    if (c == 0) Win[r * 4 + 3] = 0.f;
  } else if (c < 3 + FC) {
    v = feats[((size_t)b * N + j) * FC + (c - 3)];
  } else {
    v = 0.f;
  }
  A[(size_t)r * lda + c] = v;
}

// group_all build (SA3): rows r = b*Kk + k
__global__ __launch_bounds__(256) void k_buildall(const float* __restrict__ xyz,
                                                  const float* __restrict__ feats, int FC,
                                                  float* __restrict__ A, int lda,
                                                  float* __restrict__ Win, long total) {
  long idx = (long)blockIdx.x * 256 + threadIdx.x;
  if (idx >= total) return;
  long r = idx / lda;
  int c = (int)(idx % lda);
  float v;
  if (c < 3) {
    v = xyz[r * 3 + c];
    Win[r * 4 + c] = v;
    if (c == 0) Win[r * 4 + 3] = 0.f;
  } else if (c < 3 + FC) {
    v = feats[(size_t)r * FC + (c - 3)];
  } else {
    v = 0.f;
  }
  A[(size_t)r * lda + c] = v;
}

// ratio for group_all: per batch max over Kk=64
__global__ __launch_bounds__(64) void k_ratioall(const float* __restrict__ invden,
                                                 float* __restrict__ ratio) {
  __shared__ float sm[64];
  int b = blockIdx.x, tid = threadIdx.x;
  float v = invden[b * 64 + tid];
  sm[tid] = v;
  __syncthreads();
  for (int s = 32; s > 0; s >>= 1) {
    if (tid < s) sm[tid] = fmaxf(sm[tid], sm[tid + s]);
    __syncthreads();
  }
  ratio[b * 64 + tid] = v / sm[0];
}

// ---------------------------------------------------------------------------
// fp32 WMMA GEMM: C[M,N] = A[M,K] * W[N,K]^T (+bias). One wave = one 16x16 tile.
// V_WMMA_F32_16X16X4_F32 fragment layout:
//   A: lane<16 -> row m=l16, k=kstep+{0,1}; lane>=16 -> k=kstep+{2,3}
//   B: lane<16 -> col n=l16, k=kstep+{0,1}; lane>=16 -> k=kstep+{2,3}
//   D: lane l holds col n=l16, rows (l<16 ? 0..7 : 8..15)
// ---------------------------------------------------------------------------
__global__ __launch_bounds__(32) void k_gemm_wmma(const float* __restrict__ A, int lda,
                                                  const float* __restrict__ W, int ldw,
                                                  float* __restrict__ C, int ldc, int K,
                                                  const float* __restrict__ bias) {
  int lane = threadIdx.x;
  int half = lane >> 4, l16 = lane & 15;
  int m = blockIdx.x * 16 + l16;
  int n = blockIdx.y * 16 + l16;
  const float* ap = A + (size_t)m * lda + half * 2;
  const float* wp = W + (size_t)n * ldw + half * 2;
  v8f acc = {0.f, 0.f, 0.f, 0.f, 0.f, 0.f, 0.f, 0.f};
#pragma unroll 2
  for (int k = 0; k < K; k += 4) {
    v2f a = *(const v2f*)(ap + k);
    v2f b = *(const v2f*)(wp + k);
    acc = __builtin_amdgcn_wmma_f32_16x16x4_f32(false, a, false, b, (short)0, acc, false, false);
  }
  int col = blockIdx.y * 16 + l16;
  float bv = bias ? bias[col] : 0.f;
  int row0 = blockIdx.x * 16 + half * 8;
#pragma unroll
  for (int r = 0; r < 8; ++r)
    C[(size_t)(row0 + r) * ldc + col] = acc[r] + bv;
}

// feat[bs][(c)*16+w] = sum_k P[bs,k,c] * Wn[bs,k,w]   (M=C, N=16, K=ns)
__global__ __launch_bounds__(32) void k_feat_wmma(const float* __restrict__ P, int ldp,
                                                  const float* __restrict__ Wn,
                                                  float* __restrict__ F, int ns, int C) {
  int bs = blockIdx.x;
  int lane = threadIdx.x, half = lane >> 4, l16 = lane & 15;
  int c = blockIdx.y * 16 + l16;
  const float* pb = P + (size_t)bs * ns * ldp;
  const float* wb = Wn + (size_t)bs * ns * 16;
  v8f acc = {0.f, 0.f, 0.f, 0.f, 0.f, 0.f, 0.f, 0.f};
  for (int k = 0; k < ns; k += 4) {
    int k0 = k + half * 2;
    v2f a, b;
    a.x = pb[(size_t)k0 * ldp + c];
    a.y = pb[(size_t)(k0 + 1) * ldp + c];
    b.x = wb[k0 * 16 + l16];
    b.y = wb[(k0 + 1) * 16 + l16];
    acc = __builtin_amdgcn_wmma_f32_16x16x4_f32(false, a, false, b, (short)0, acc, false, false);
  }
  int c0 = blockIdx.y * 16 + half * 8;
#pragma unroll
  for (int r = 0; r < 8; ++r)
    F[(size_t)bs * C * 16 + (size_t)(c0 + r) * 16 + l16] = acc[r];
}

// ---------------------------------------------------------------------------
// batch-norm stats per channel over R rows (population variance)
// ---------------------------------------------------------------------------
__global__ __launch_bounds__(256) void k_bn_stats(const float* __restrict__ X, int ldx,
                                                  int R, int C, float* __restrict__ stats) {
  __shared__ float s1[256], s2[256];
  int c = blockIdx.x, tid = threadIdx.x;
  float a = 0.f, b = 0.f;
  for (int r = tid; r < R; r += 256) {
    float v = X[(size_t)r * ldx + c];
    a += v;
    b += v * v;
  }
  s1[tid] = a;
  s2[tid] = b;
  __syncthreads();
  for (int s = 128; s > 0; s >>= 1) {
    if (tid < s) { s1[tid] += s1[tid + s]; s2[tid] += s2[tid + s]; }
    __syncthreads();
  }
  if (tid == 0) {
    float mu = s1[0] / (float)R;
    stats[c] = mu;
    stats[C + c] = s2[0] / (float)R - mu * mu;
  }
}

// y = relu(g*(x-mu)*rsqrt(var+eps)+be) [* ratio[row]]
__global__ __launch_bounds__(256) void k_bn_apply(float* __restrict__ X, int ldx, long R,
                                                  int C, const float* __restrict__ stats,
                                                  const float* __restrict__ g,
                                                  const float* __restrict__ be,
                                                  const float* __restrict__ ratio) {
  long idx = (long)blockIdx.x * 256 + threadIdx.x;
  if (idx >= R * (long)C) return;
  long r = idx / C;
  int c = (int)(idx % C);
  float mu = stats[c], var = stats[C + c];
  float y = g[c] * (X[(size_t)r * ldx + c] - mu) * rsqrtf(var + EPSV) + be[c];
  y = fmaxf(y, 0.f);
  if (ratio) y *= ratio[r];
  X[(size_t)r * ldx + c] = y;
}

// per-row group norm over C channels + relu (C <= 256)
__global__ __launch_bounds__(256) void k_gn_relu(float* __restrict__ X, int C,
                                                 const float* __restrict__ g,
                                                 const float* __restrict__ be) {
  __shared__ float s1[256], s2[256];
  int r = blockIdx.x, tid = threadIdx.x;
  float v = (tid < C) ? X[(size_t)r * C + tid] : 0.f;
  s1[tid] = v;
  s2[tid] = v * v;
  __syncthreads();
  for (int s = 128; s > 0; s >>= 1) {
    if (tid < s) { s1[tid] += s1[tid + s]; s2[tid] += s2[tid + s]; }
    __syncthreads();
  }
  float mu = s1[0] / (float)C;
  float var = s2[0] / (float)C - mu * mu;
  if (tid < C)
    X[(size_t)r * C + tid] = fmaxf(g[tid] * (v - mu) * rsqrtf(var + EPSV) + be[tid], 0.f);
}

// misc small kernels
__global__ __launch_bounds__(256) void k_sub(const float* __restrict__ a,
                                             const float* __restrict__ b,
                                             float* __restrict__ o, int n) {
  int i = blockIdx.x * 256 + threadIdx.x;
  if (i < n) o[i] = a[i] - b[i];
}
__global__ __launch_bounds__(64) void k_copyout(const float* __restrict__ t,
                                                float* __restrict__ o) {
  int i = threadIdx.x;
  if (i < 48) o[i] = t[(i / 3) * 16 + (i % 3)];
}
__global__ __launch_bounds__(256) void k_pack2d(const float* __restrict__ src, int srows,
                                                int scols, float* __restrict__ dst,
                                                int dcols, int total) {
  int idx = blockIdx.x * 256 + threadIdx.x;
  if (idx >= total) return;
  int r = idx / dcols, c = idx % dcols;
  dst[idx] = (r < srows && c < scols) ? src[r * scols + c] : 0.f;
}
__global__ __launch_bounds__(64) void k_pack1d(const float* __restrict__ src, int n,
                                               float* __restrict__ dst, int dn) {
  int i = threadIdx.x;
  if (i < dn) dst[i] = (i < n) ? src[i] : 0.f;
}

// ---------------------------------------------------------------------------
extern "C" void kernel_launch(void* const* d_in, const int* in_sizes, int n_in,
                              void* d_out, int out_size, void* d_ws, size_t ws_size,
                              hipStream_t stream) {
  // Input ordering: top-level dict insertion order {xyz, xyz_goal, params},
  // params flattened as a jax pytree (sorted keys). Auto-detect the alternative
  // (params-first, fully sorted) via in_sizes[0].
  int ixyz = 0, igoal = 1, pbase = 2;
  if (in_sizes[0] != 16 * 6 * 2048) { pbase = 0; ixyz = n_in - 2; igoal = n_in - 1; }

  const float* xyzX = (const float*)d_in[ixyz];
  const float* xyzG = (const float*)d_in[igoal];
  const float* fc1W = (const float*)d_in[pbase + 0];
  const float* fc1b = (const float*)d_in[pbase + 1];
  const float* fc3W = (const float*)d_in[pbase + 2];
  const float* fc3b = (const float*)d_in[pbase + 3];
  const float* fc4W = (const float*)d_in[pbase + 4];
  const float* fc4b = (const float*)d_in[pbase + 5];
  const float* fc5W = (const float*)d_in[pbase + 6];
  const float* fc5b = (const float*)d_in[pbase + 7];
  const float* gn1g = (const float*)d_in[pbase + 8];
  const float* gn1b = (const float*)d_in[pbase + 9];
  const float* gn3g = (const float*)d_in[pbase + 10];
  const float* gn3b = (const float*)d_in[pbase + 11];
  const float* gn4g = (const float*)d_in[pbase + 12];
  const float* gn4b = (const float*)d_in[pbase + 13];
  // sa leaves (sorted keys: bnlin, lin, mlp, wn):
  // 0 bnlin.g, 1 bnlin.be, 2 lin.W, 3 lin.b, 4 mlp.W, 5 mlp.b, 6 mlp.g, 7 mlp.be,
  // 8+4j wn_j.{W,b,g,be}
  auto SA = [&](int i, int leaf) { return (const float*)d_in[pbase + 14 + 20 * i + leaf]; };

  // workspace bump allocator (floats, 256B aligned chunks)
  float* wsp = (float*)d_ws;
  size_t off = 0;
  auto alloc = [&](size_t nflt) {
    float* p = wsp + off;
    off += (nflt + 63) & ~(size_t)63;
    return p;
  };

  // packed weights
  float* mlp1p = alloc(128 * 12);
  float* mlp2p = alloc(256 * 132);
  float* mlp3p = alloc(512 * 260);
  float* wnWp[3][3];
  float* wnGp[3][3];
  float* wnBp[3][3];
  for (int i = 0; i < 3; ++i) {
    wnWp[i][0] = alloc(16 * 4);
    wnWp[i][1] = alloc(16 * 16);
    wnWp[i][2] = alloc(16 * 16);
    for (int j = 0; j < 3; ++j) { wnGp[i][j] = alloc(16); wnBp[i][j] = alloc(16); }
  }
  float* fc5Wp = alloc(16 * 32);
  float* fc5bp = alloc(16);

  // data buffers (re-used across the two encodes)
  float* pts = alloc(16 * 2048 * 6);
  float* xyz0 = alloc(16 * 2048 * 3);
  float* invden = alloc(16 * 2048);
  int* fpsi = (int*)alloc(2048);
  int* knni = (int*)alloc(16384);
  float* nx1 = alloc(16 * 128 * 3);
  float* nx2 = alloc(16 * 64 * 3);
  float* ratio = alloc(16384);
  float* WinB = alloc(16384 * 4);
  float* wnA = alloc(16384 * 16);
  float* wnBf = alloc(16384 * 16);
  float* stats = alloc(1024);
  float* Abuf = alloc(16384 * 132);   // max over A1(12), A2(132), A3(260 w/ 1024 rows)
  float* Mbuf = alloc(16384 * 256);   // max over M1, M2, M3
  float* Fbuf = alloc(1024 * 4096);   // max over feat1/2/3
  float* l1pts = alloc(2048 * 128);
  float* l2pts = alloc(1024 * 256);
  float* l3x = alloc(16 * 512);
  float* l3g = alloc(16 * 512);
  float* hbuf = alloc(16 * 512);
  float* h1 = alloc(16 * 256);
  float* h2 = alloc(16 * 128);
  float* h3 = alloc(16 * 32);
  float* otmp = alloc(16 * 16);
  (void)ws_size; (void)out_size;

  // ---- pack weights (zero-padded to WMMA-friendly shapes) ----
  k_pack2d<<<(128 * 12 + 255) / 256, 256, 0, stream>>>(SA(0, 4), 128, 9, mlp1p, 12, 128 * 12);
  k_pack2d<<<(256 * 132 + 255) / 256, 256, 0, stream>>>(SA(1, 4), 256, 131, mlp2p, 132, 256 * 132);
  k_pack2d<<<(512 * 260 + 255) / 256, 256, 0, stream>>>(SA(2, 4), 512, 259, mlp3p, 260, 512 * 260);
  for (int i = 0; i < 3; ++i) {
    k_pack2d<<<1, 256, 0, stream>>>(SA(i, 8), 8, 3, wnWp[i][0], 4, 64);
    k_pack2d<<<1, 256, 0, stream>>>(SA(i, 12), 8, 8, wnWp[i][1], 16, 256);
    k_pack2d<<<1, 256, 0, stream>>>(SA(i, 16), 16, 8, wnWp[i][2], 16, 256);
    int odim[3] = {8, 8, 16};
    for (int j = 0; j < 3; ++j) {
      k_pack1d<<<1, 64, 0, stream>>>(SA(i, 8 + 4 * j + 2), odim[j], wnGp[i][j], 16);
      k_pack1d<<<1, 64, 0, stream>>>(SA(i, 8 + 4 * j + 3), odim[j], wnBp[i][j], 16);
    }
  }
  k_pack2d<<<(16 * 32 + 255) / 256, 256, 0, stream>>>(fc5W, 3, 32, fc5Wp, 32, 16 * 32);
  k_pack1d<<<1, 64, 0, stream>>>(fc5b, 3, fc5bp, 16);

  // weight-net chain: Win(R x 4) -> 16 -> 16 -> 16, BN+relu each stage; result in wnA
  auto wn_chain = [&](int i, float* Win, int R) {
    k_gemm_wmma<<<dim3(R / 16, 1), 32, 0, stream>>>(Win, 4, wnWp[i][0], 4, wnA, 16, 4, nullptr);
    k_bn_stats<<<16, 256, 0, stream>>>(wnA, 16, R, 16, stats);
    k_bn_apply<<<(R * 16 + 255) / 256, 256, 0, stream>>>(wnA, 16, R, 16, stats, wnGp[i][0], wnBp[i][0], nullptr);
    k_gemm_wmma<<<dim3(R / 16, 1), 32, 0, stream>>>(wnA, 16, wnWp[i][1], 16, wnBf, 16, 16, nullptr);
    k_bn_stats<<<16, 256, 0, stream>>>(wnBf, 16, R, 16, stats);
    k_bn_apply<<<(R * 16 + 255) / 256, 256, 0, stream>>>(wnBf, 16, R, 16, stats, wnGp[i][1], wnBp[i][1], nullptr);
    k_gemm_wmma<<<dim3(R / 16, 1), 32, 0, stream>>>(wnBf, 16, wnWp[i][2], 16, wnA, 16, 16, nullptr);
    k_bn_stats<<<16, 256, 0, stream>>>(wnA, 16, R, 16, stats);
    k_bn_apply<<<(R * 16 + 255) / 256, 256, 0, stream>>>(wnA, 16, R, 16, stats, wnGp[i][2], wnBp[i][2], nullptr);
  };

  auto encode = [&](const float* xin, float* l3) {
    // ---------------- SA1: N=2048 -> S=128, ns=8, 9->128 ----------------
    k_transpose<<<768, 256, 0, stream>>>(xin, pts, xyz0);
    k_density<<<dim3(16, 8), 256, 0, stream>>>(xyz0, invden, 2048, 0.1f);
    k_fps<<<16, 256, 0, stream>>>(xyz0, 2048, 128, fpsi);
    k_knn<8><<<8, 256, 0, stream>>>(xyz0, 2048, fpsi, 128, invden, nx1, knni, ratio, 2048);
    k_gather<<<(16384 * 12 + 255) / 256, 256, 0, stream>>>(xyz0, 2048, pts, 6, nx1, 128, 8, knni,
                                                           Abuf, 12, WinB, (long)16384 * 12);
    k_gemm_wmma<<<dim3(1024, 8), 32, 0, stream>>>(Abuf, 12, mlp1p, 12, Mbuf, 128, 12, nullptr);
    k_bn_stats<<<128, 256, 0, stream>>>(Mbuf, 128, 16384, 128, stats);
    k_bn_apply<<<(16384 * 128 + 255) / 256, 256, 0, stream>>>(Mbuf, 128, 16384, 128, stats,
                                                              SA(0, 6), SA(0, 7), ratio);
    wn_chain(0, WinB, 16384);
    k_feat_wmma<<<dim3(2048, 8), 32, 0, stream>>>(Mbuf, 128, wnA, Fbuf, 8, 128);
    k_gemm_wmma<<<dim3(128, 8), 32, 0, stream>>>(Fbuf, 2048, SA(0, 2), 2048, l1pts, 128, 2048, nullptr);
    k_bn_stats<<<128, 256, 0, stream>>>(l1pts, 128, 2048, 128, stats);
    k_bn_apply<<<(2048 * 128 + 255) / 256, 256, 0, stream>>>(l1pts, 128, 2048, 128, stats,
                                                             SA(0, 0), SA(0, 1), nullptr);
    // ---------------- SA2: N=128 -> S=64, ns=16, 131->256 ----------------
    k_density<<<dim3(16, 1), 256, 0, stream>>>(nx1, invden, 128, 0.2f);
    k_fps<<<16, 256, 0, stream>>>(nx1, 128, 64, fpsi);
    k_knn<16><<<4, 256, 0, stream>>>(nx1, 128, fpsi, 64, invden, nx2, knni, ratio, 1024);
    k_gather<<<(16384 * 132 + 255) / 256, 256, 0, stream>>>(nx1, 128, l1pts, 128, nx2, 64, 16,
                                                            knni, Abuf, 132, WinB, (long)16384 * 132);
    k_gemm_wmma<<<dim3(1024, 16), 32, 0, stream>>>(Abuf, 132, mlp2p, 132, Mbuf, 256, 132, nullptr);
    k_bn_stats<<<256, 256, 0, stream>>>(Mbuf, 256, 16384, 256, stats);
    k_bn_apply<<<(16384 * 256 + 255) / 256, 256, 0, stream>>>(Mbuf, 256, 16384, 256, stats,
                                                              SA(1, 6), SA(1, 7), ratio);
    wn_chain(1, WinB, 16384);
    k_feat_wmma<<<dim3(1024, 16), 32, 0, stream>>>(Mbuf, 256, wnA, Fbuf, 16, 256);
    k_gemm_wmma<<<dim3(64, 16), 32, 0, stream>>>(Fbuf, 4096, SA(1, 2), 4096, l2pts, 256, 4096, nullptr);
    k_bn_stats<<<256, 256, 0, stream>>>(l2pts, 256, 1024, 256, stats);
    k_bn_apply<<<(1024 * 256 + 255) / 256, 256, 0, stream>>>(l2pts, 256, 1024, 256, stats,
                                                             SA(1, 0), SA(1, 1), nullptr);
    // ---------------- SA3: group_all, K=64, 259->512 ----------------
    k_density<<<dim3(16, 1), 256, 0, stream>>>(nx2, invden, 64, 0.4f);
    k_ratioall<<<16, 64, 0, stream>>>(invden, ratio);
    k_buildall<<<(1024 * 260 + 255) / 256, 256, 0, stream>>>(nx2, l2pts, 256, Abuf, 260, WinB,
                                                             (long)1024 * 260);
    k_gemm_wmma<<<dim3(64, 32), 32, 0, stream>>>(Abuf, 260, mlp3p, 260, Mbuf, 512, 260, nullptr);
    k_bn_stats<<<512, 256, 0, stream>>>(Mbuf, 512, 1024, 512, stats);
    k_bn_apply<<<(1024 * 512 + 255) / 256, 256, 0, stream>>>(Mbuf, 512, 1024, 512, stats,
                                                             SA(2, 6), SA(2, 7), ratio);
    wn_chain(2, WinB, 1024);
    k_feat_wmma<<<dim3(16, 32), 32, 0, stream>>>(Mbuf, 512, wnA, Fbuf, 64, 512);
    k_gemm_wmma<<<dim3(1, 32), 32, 0, stream>>>(Fbuf, 8192, SA(2, 2), 8192, l3, 512, 8192, nullptr);
    k_bn_stats<<<512, 256, 0, stream>>>(l3, 512, 16, 512, stats);
    k_bn_apply<<<(16 * 512 + 255) / 256, 256, 0, stream>>>(l3, 512, 16, 512, stats,
                                                           SA(2, 0), SA(2, 1), nullptr);
  };

  encode(xyzX, l3x);
  encode(xyzG, l3g);

  // ---------------- head ----------------
  k_sub<<<(8192 + 255) / 256, 256, 0, stream>>>(l3g, l3x, hbuf, 8192);
  k_gemm_wmma<<<dim3(1, 16), 32, 0, stream>>>(hbuf, 512, fc1W, 512, h1, 256, 512, fc1b);
  k_gn_relu<<<16, 256, 0, stream>>>(h1, 256, gn1g, gn1b);
  k_gemm_wmma<<<dim3(1, 8), 32, 0, stream>>>(h1, 256, fc3W, 256, h2, 128, 256, fc3b);
  k_gn_relu<<<16, 256, 0, stream>>>(h2, 128, gn3g, gn3b);
  k_gemm_wmma<<<dim3(1, 2), 32, 0, stream>>>(h2, 128, fc4W, 128, h3, 32, 128, fc4b);
  k_gn_relu<<<16, 256, 0, stream>>>(h3, 32, gn4g, gn4b);
  k_gemm_wmma<<<dim3(1, 1), 32, 0, stream>>>(h3, 32, fc5Wp, 32, otmp, 16, 32, fc5bp);
  k_copyout<<<1, 64, 0, stream>>>(otmp, (float*)d_out);
}